// YoloV5Loss_13743895347410
// MI455X (gfx1250) — compile-verified
//
#include <hip/hip_runtime.h>
#include <math.h>

#define EPSF 1e-7f

typedef __attribute__((ext_vector_type(2))) float v2f;
typedef __attribute__((ext_vector_type(8))) float v8f;

// ANCHORS[level][a] = raw anchor / stride
__constant__ float c_anch[3][3][2] = {
  {{1.25f, 1.625f}, {2.0f, 3.75f}, {4.125f, 2.875f}},
  {{1.875f, 3.8125f}, {3.875f, 2.8125f}, {3.6875f, 7.4375f}},
  {{3.625f, 2.8125f}, {4.875f, 6.1875f}, {11.65625f, 10.1875f}}
};
__constant__ float c_off[5][2] = {{0.f,0.f},{1.f,0.f},{0.f,1.f},{-1.f,0.f},{0.f,-1.f}};

__device__ __forceinline__ float bce_f(float x, float t) {
  return fmaxf(x, 0.f) - x * t + log1pf(expf(-fabsf(x)));
}
__device__ __forceinline__ float sigmoid_f(float x) { return 1.f / (1.f + expf(-x)); }

// ---------------- stage 0: zero workspace ----------------
__global__ void zero_kernel(float* __restrict__ ws, int n) {
  int i = blockIdx.x * blockDim.x + threadIdx.x;
  int stride = gridDim.x * blockDim.x;
  for (; i < n; i += stride) ws[i] = 0.f;
}

// ---------------- stage 1: matched-target losses ----------------
// acc layout (floats): [0..2] box_sum, [3..5] cls_sum, [6..8] obj_sum, [9..11] match count
// (compiler's atomic optimizer aggregates same-address atomicAdd across the wave,
//  and there are only 4800 threads/level, so per-thread atomics are fine here)
__global__ void match_kernel(const float* __restrict__ pred,
                             const float* __restrict__ ann,
                             float* __restrict__ acc,
                             float* __restrict__ tobj,
                             int level, int H, int W, float stride_px) {
  const int B = 16, N = 20, A = 3, O = 5;
  int tid = blockIdx.x * blockDim.x + threadIdx.x;
  if (tid >= B * N * A * O) return;
  int o = tid % O;
  int a = (tid / O) % A;
  int n = (tid / (O * A)) % N;
  int b = tid / (O * A * N);

  const float* r = ann + (b * N + n) * 5;
  float x1 = r[0], y1 = r[1], x2 = r[2], y2 = r[3], cls = r[4];
  if (!(cls >= 0.f)) return;  // valid

  float inv_s = 1.f / stride_px;
  float gx = (x1 + x2) * 0.5f * inv_s;
  float gy = (y1 + y2) * 0.5f * inv_s;
  float gw = (x2 - x1) * inv_s;
  float gh = (y2 - y1) * inv_s;

  float aw = c_anch[level][a][0], ah = c_anch[level][a][1];
  float rw = gw / aw, rh = gh / ah;
  float rmax = fmaxf(fmaxf(rw, 1.f / (rw + EPSF)), fmaxf(rh, 1.f / (rh + EPSF)));
  if (!(rmax < 4.f)) return;  // ratio_ok (ANCHOR_T)

  float fx = gx - floorf(gx), fy = gy - floorf(gy);
  bool ok;
  if (o == 0)      ok = true;
  else if (o == 1) ok = (fx < 0.5f) && (gx > 1.f);
  else if (o == 2) ok = (fy < 0.5f) && (gy > 1.f);
  else if (o == 3) ok = (fx > 0.5f) && (gx < (float)W - 1.f);
  else             ok = (fy > 0.5f) && (gy < (float)H - 1.f);
  if (!ok) return;

  float gif = fminf(fmaxf(floorf(gx - c_off[o][0] * 0.5f), 0.f), (float)(W - 1));
  float gjf = fminf(fmaxf(floorf(gy - c_off[o][1] * 0.5f), 0.f), (float)(H - 1));
  int gi = (int)gif, gj = (int)gjf;

  const float* p = pred + (size_t)((((b * H + gj) * W + gi) * A) + a) * 85;

  float px = sigmoid_f(p[0]) * 2.f - 0.5f;
  float py = sigmoid_f(p[1]) * 2.f - 0.5f;
  float sw = sigmoid_f(p[2]) * 2.f;
  float sh = sigmoid_f(p[3]) * 2.f;
  float pw = sw * sw * aw;
  float ph = sh * sh * ah;
  float tx = gx - gif, ty = gy - gjf;

  // CIoU(pbox, tbox)
  float px1 = px - pw * 0.5f, px2 = px + pw * 0.5f;
  float py1 = py - ph * 0.5f, py2 = py + ph * 0.5f;
  float tx1 = tx - gw * 0.5f, tx2 = tx + gw * 0.5f;
  float ty1 = ty - gh * 0.5f, ty2 = ty + gh * 0.5f;
  float iw = fmaxf(fminf(px2, tx2) - fmaxf(px1, tx1), 0.f);
  float ih = fmaxf(fminf(py2, ty2) - fmaxf(py1, ty1), 0.f);
  float inter = iw * ih;
  float uni = pw * ph + gw * gh - inter + EPSF;
  float iou = inter / uni;
  float cw = fmaxf(px2, tx2) - fminf(px1, tx1);
  float ch = fmaxf(py2, ty2) - fminf(py1, ty1);
  float c2 = cw * cw + ch * ch + EPSF;
  float rho2 = (px - tx) * (px - tx) + (py - ty) * (py - ty);
  float da = atanf(gw / (gh + EPSF)) - atanf(pw / (ph + EPSF));
  const float PI = 3.14159265358979323846f;
  float v = (4.f / (PI * PI)) * da * da;
  float alpha = v / (1.f - iou + v + EPSF);
  float ciou = iou - rho2 / c2 - alpha * v;

  atomicAdd(&acc[0 + level], 1.f - ciou);   // box sum
  int c = (int)fminf(fmaxf(cls, 0.f), 79.f);
  float cs = 0.f;
  for (int k = 0; k < 80; ++k) cs += bce_f(p[5 + k], (k == c) ? 1.f : 0.f);
  atomicAdd(&acc[3 + level], cs);           // cls sum
  atomicAdd(&acc[9 + level], 1.f);          // match count

  float iou_d = fminf(fmaxf(ciou, 0.f), 1.f);  // non-negative -> int-bits max == float max
  atomicMax(reinterpret_cast<int*>(&tobj[(size_t)((b * H + gj) * W + gi) * A + a]),
            __float_as_int(iou_d));
}

// ---------------- stage 2: objectness BCE reduction (bulk of HBM traffic) ----------------
// Each thread handles OBJ_ITER strided elements (prefetch warms the next chunk).
// Wave reduction via v_wmma_f32_16x16x4_f32:
//   A(16x4): lane L<16 -> a[L][0]=s; lane L>=16 -> a[L-16][2]=s. B(4x16)=ones.
//   D[m][n] = s_m + s_{m+16}; summing the 8 D VGPRs gives rows 0..7 (lanes 0-15)
//   / rows 8..15 (lanes 16-31); shfl(lane0)+shfl(lane16) totals the wave.
// Then an LDS block reduction -> ONE atomicAdd per block (kills same-address
// atomic serialization that would otherwise rival the streaming time).
#define OBJ_ITER 4

__global__ void obj_kernel(const float* __restrict__ pred,
                           const float* __restrict__ tobj,
                           float* __restrict__ acc, int level, int M) {
  __shared__ float part[8];  // 256 threads = 8 waves
  int tid = blockIdx.x * blockDim.x + threadIdx.x;
  int stride = gridDim.x * blockDim.x;
  float s = 0.f;
  for (int i = tid; i < M; i += stride) {
    if (i + stride < M)
      __builtin_prefetch(&pred[(size_t)(i + stride) * 85 + 4], 0, 1);  // global_prefetch_b8
    float x = pred[(size_t)i * 85 + 4];
    s += bce_f(x, tobj[i]);
  }

  v2f a;  a[0]  = s;   a[1]  = 0.f;
  v2f bm; bm[0] = 1.f; bm[1] = 1.f;
  v8f cacc = {};
  cacc = __builtin_amdgcn_wmma_f32_16x16x4_f32(
      /*neg_a=*/false, a, /*neg_b=*/false, bm,
      /*c_mod=*/(short)0, cacc, /*reuse_a=*/false, /*reuse_b=*/false);
  float t = cacc[0] + cacc[1] + cacc[2] + cacc[3] + cacc[4] + cacc[5] + cacc[6] + cacc[7];
  float lo = __shfl(t, 0, 32);
  float hi = __shfl(t, 16, 32);
  float wave_sum = lo + hi;

  int wave = threadIdx.x >> 5;
  int lane = threadIdx.x & 31;
  if (lane == 0) part[wave] = wave_sum;
  __syncthreads();
  if (threadIdx.x == 0) {
    float bs = part[0] + part[1] + part[2] + part[3]
             + part[4] + part[5] + part[6] + part[7];
    atomicAdd(&acc[6 + level], bs);
  }
}

// ---------------- stage 3: combine ----------------
__global__ void final_kernel(const float* __restrict__ acc, float* __restrict__ out) {
  const float lvlw[3] = {4.f, 1.f, 0.4f};
  const float Ms[3]   = {307200.f, 76800.f, 19200.f};  // B*H*W*A per level
  float box = 0.f, objl = 0.f, clsl = 0.f;
  for (int l = 0; l < 3; ++l) {
    float nm = fmaxf(acc[9 + l], 1.f);
    box  += acc[0 + l] / nm;
    clsl += acc[3 + l] / (nm * 80.f);
    objl += (acc[6 + l] / Ms[l]) * lvlw[l];
  }
  out[0] = 0.05f * box + 1.0f * objl + 0.5f * clsl;
}

extern "C" void kernel_launch(void* const* d_in, const int* in_sizes, int n_in,
                              void* d_out, int out_size, void* d_ws, size_t ws_size,
                              hipStream_t stream) {
  const float* p0  = (const float*)d_in[0];  // (16,80,80,3,85)
  const float* p1  = (const float*)d_in[1];  // (16,40,40,3,85)
  const float* p2  = (const float*)d_in[2];  // (16,20,20,3,85)
  const float* ann = (const float*)d_in[3];  // (16,20,5)

  float* ws    = (float*)d_ws;
  float* acc   = ws;             // 16 floats header
  float* tobj0 = ws + 16;        // 307200
  float* tobj1 = tobj0 + 307200; // 76800
  float* tobj2 = tobj1 + 76800;  // 19200
  const int M0 = 307200, M1 = 76800, M2 = 19200;
  const int total_zero = 16 + M0 + M1 + M2;

  zero_kernel<<<512, 256, 0, stream>>>(ws, total_zero);

  const int match_threads = 16 * 20 * 3 * 5;  // 4800
  dim3 mg((match_threads + 255) / 256);
  match_kernel<<<mg, 256, 0, stream>>>(p0, ann, acc, tobj0, 0, 80, 80, 8.f);
  match_kernel<<<mg, 256, 0, stream>>>(p1, ann, acc, tobj1, 1, 40, 40, 16.f);
  match_kernel<<<mg, 256, 0, stream>>>(p2, ann, acc, tobj2, 2, 20, 20, 32.f);

  const int chunk = 256 * OBJ_ITER;
  obj_kernel<<<(M0 + chunk - 1) / chunk, 256, 0, stream>>>(p0, tobj0, acc, 0, M0);
  obj_kernel<<<(M1 + chunk - 1) / chunk, 256, 0, stream>>>(p1, tobj1, acc, 1, M1);
  obj_kernel<<<(M2 + chunk - 1) / chunk, 256, 0, stream>>>(p2, tobj2, acc, 2, M2);

  final_kernel<<<1, 1, 0, stream>>>(acc, (float*)d_out);
}